// My_loss_30331059045011
// MI455X (gfx1250) — compile-verified
//
#include <hip/hip_runtime.h>

typedef float v4f __attribute__((ext_vector_type(4)));

#define D_CLASSES 256
#define WAVES_PER_BLOCK 8
#define BLOCK_THREADS (WAVES_PER_BLOCK * 32)
#define PF_ROWS 64   // prefetch distance: 64 rows = 64 KiB per stream

__device__ __forceinline__ float wave_reduce_max(float v) {
#pragma unroll
    for (int off = 16; off > 0; off >>= 1)
        v = fmaxf(v, __shfl_xor(v, off, 32));
    return v;
}

__device__ __forceinline__ float wave_reduce_sum(float v) {
#pragma unroll
    for (int off = 16; off > 0; off >>= 1)
        v += __shfl_xor(v, off, 32);
    return v;
}

// One wave32 per row: softmax over 256 classes fused with row-dot against p0.
// Each lane holds 8 consecutive floats -> two global_load_b128 (NT) per stream.
__global__ __launch_bounds__(BLOCK_THREADS) void rowloss_kernel(
    const float* __restrict__ z, const float* __restrict__ p,
    float* __restrict__ partials, int nRows)
{
    const int lane = threadIdx.x & 31;
    const int wave = threadIdx.x >> 5;
    const int row  = blockIdx.x * WAVES_PER_BLOCK + wave;
    __shared__ float acc[WAVES_PER_BLOCK];

    float contrib = 0.0f;
    if (row < nRows) {
        const size_t base = (size_t)row * D_CLASSES + (size_t)lane * 8;

        // gfx1250 speculative prefetch (global_prefetch_b8, GL2 scope): pull the
        // stream PF_ROWS ahead. Speculative => OOB at stream tail is silently
        // dropped by hardware, so no guard branch / exec-mask dance needed.
        __builtin_prefetch(z + base + (size_t)PF_ROWS * D_CLASSES, 0, 1);
        __builtin_prefetch(p + base + (size_t)PF_ROWS * D_CLASSES, 0, 1);

        // Touch-once streaming: non-temporal B128 loads.
        v4f z0 = __builtin_nontemporal_load((v4f*)(z + base));
        v4f z1 = __builtin_nontemporal_load((v4f*)(z + base) + 1);
        v4f q0 = __builtin_nontemporal_load((v4f*)(p + base));
        v4f q1 = __builtin_nontemporal_load((v4f*)(p + base) + 1);

        // Row max (numerical stability, matches jax.nn.softmax).
        float m = fmaxf(fmaxf(fmaxf(z0.x, z0.y), fmaxf(z0.z, z0.w)),
                        fmaxf(fmaxf(z1.x, z1.y), fmaxf(z1.z, z1.w)));
        m = wave_reduce_max(m);

        float e0 = __expf(z0.x - m), e1 = __expf(z0.y - m);
        float e2 = __expf(z0.z - m), e3 = __expf(z0.w - m);
        float e4 = __expf(z1.x - m), e5 = __expf(z1.y - m);
        float e6 = __expf(z1.z - m), e7 = __expf(z1.w - m);

        float s = ((e0 + e1) + (e2 + e3)) + ((e4 + e5) + (e6 + e7));
        float d = ((e0 * q0.x + e1 * q0.y) + (e2 * q0.z + e3 * q0.w)) +
                  ((e4 * q1.x + e5 * q1.y) + (e6 * q1.z + e7 * q1.w));
        s = wave_reduce_sum(s);
        d = wave_reduce_sum(d);

        float res = d / s;                       // sum(softmax(z) * p0) for this row
        contrib = 2.0f * (1.0f - sqrtf(res));    // (1 - res^0.5) / 0.5
    }

    if (lane == 0) acc[wave] = contrib;
    __syncthreads();
    if (threadIdx.x == 0) {
        float t = 0.0f;
#pragma unroll
        for (int i = 0; i < WAVES_PER_BLOCK; ++i) t += acc[i];
        partials[blockIdx.x] = t;
    }
}

// Deterministic final reduction of per-block partials -> scalar mean.
__global__ __launch_bounds__(1024) void reduce_kernel(
    const float* __restrict__ partials, float* __restrict__ out,
    int n, float invN)
{
    __shared__ float sm[1024];
    float s = 0.0f;
    for (int i = threadIdx.x; i < n; i += 1024) s += partials[i];
    sm[threadIdx.x] = s;
    __syncthreads();
#pragma unroll
    for (int off = 512; off > 0; off >>= 1) {
        if ((int)threadIdx.x < off) sm[threadIdx.x] += sm[threadIdx.x + off];
        __syncthreads();
    }
    if (threadIdx.x == 0) out[0] = sm[0] * invN;
}

extern "C" void kernel_launch(void* const* d_in, const int* in_sizes, int n_in,
                              void* d_out, int out_size, void* d_ws, size_t ws_size,
                              hipStream_t stream)
{
    const float* z = (const float*)d_in[0];   // zp_pro_0 [N, 256] fp32
    const float* p = (const float*)d_in[1];   // p0       [N, 256] fp32
    float* out = (float*)d_out;               // scalar fp32
    float* partials = (float*)d_ws;           // one float per block

    const int nRows  = in_sizes[0] / D_CLASSES;               // 262144
    const int blocks = (nRows + WAVES_PER_BLOCK - 1) / WAVES_PER_BLOCK;  // 32768

    rowloss_kernel<<<blocks, BLOCK_THREADS, 0, stream>>>(z, p, partials, nRows);
    reduce_kernel<<<1, 1024, 0, stream>>>(partials, out, blocks, 1.0f / (float)nRows);
}